// CausalLayer_32512902430853
// MI455X (gfx1250) — compile-verified
//
#include <hip/hip_runtime.h>

// ---------------------------------------------------------------------------
// CausalLayer for MI455X (gfx1250, wave32, WMMA).
//   sigma = mu + eps*exp(logstd)                        (elementwise, HBM-bound)
//   R = inv(I - A^T) via Neumann/Horner: X <- I + A^T X (fp32 WMMA GEMMs, L2-resident)
//   z_b = R[idx_b, idx_b] @ sigma_b                     (gathered batched WMMA matmul)
// Workspace layout (d_ws): At (4MB) | X0 (4MB) | X1 (4MB)  = 12MB total.
//
// LDS tiles are stored K-pair-interleaved (v2f per [k/2][col]) so every WMMA
// fragment is a single aligned ds_load_b64 that lands directly in an even
// VGPR pair -- no ds_load_2addr + v_mov repacking in the inner loop.
// Pair-row strides (144 / 80) are chosen so lanes 0-15 (pair p) and lanes
// 16-31 (pair p+1) read disjoint 32-bank halves: conflict-free b64 reads.
// ---------------------------------------------------------------------------

#define MATN 1024
#define NPER 128
#define NBATCH 2048
#define DDIM 64
#define SIG_ELEMS (NBATCH * NPER * DDIM)   // 16,777,216

#define ASTRIDE 144   // pairs per k-row for the 128-wide A tile (== 16 mod 32)
#define BSTRIDE 80    // pairs per k-row for the 64-wide  B tile (== 16 mod 32)

typedef __attribute__((ext_vector_type(2))) float v2f;
typedef __attribute__((ext_vector_type(8))) float v8f;

// ---------------- sigma = mu + eps * exp(logstd) ---------------------------
__global__ __launch_bounds__(256) void sigma_kernel(
    const float* __restrict__ mu, const float* __restrict__ logstd,
    const float* __restrict__ eps, float* __restrict__ sigma, int n4) {
  int i = blockIdx.x * blockDim.x + threadIdx.x;
  if (i < n4) {
    float4 m = ((const float4*)mu)[i];
    float4 l = ((const float4*)logstd)[i];
    float4 e = ((const float4*)eps)[i];
    float4 s;
    s.x = fmaf(e.x, __expf(l.x), m.x);
    s.y = fmaf(e.y, __expf(l.y), m.y);
    s.z = fmaf(e.z, __expf(l.z), m.z);
    s.w = fmaf(e.w, __expf(l.w), m.w);
    ((float4*)sigma)[i] = s;
  }
}

// ---------------- At = A^T ; X0 = I + A^T ----------------------------------
__global__ __launch_bounds__(256) void transpose_init_kernel(
    const float* __restrict__ A, float* __restrict__ At, float* __restrict__ X0) {
  __shared__ float t[32][33];
  const int bx = blockIdx.x * 32, by = blockIdx.y * 32;
  const int x = bx + threadIdx.x;
  for (int i = threadIdx.y; i < 32; i += 8)
    t[i][threadIdx.x] = A[(size_t)(by + i) * MATN + x];
  __syncthreads();
  const int ocol = by + threadIdx.x;
  for (int i = threadIdx.y; i < 32; i += 8) {
    const int orow = bx + i;
    float v = t[threadIdx.x][i];                    // = A[ocol][orow]
    At[(size_t)orow * MATN + ocol] = v;
    X0[(size_t)orow * MATN + ocol] = v + (orow == ocol ? 1.0f : 0.0f);
  }
}

// ---------------- Y = At * X (+ I)  via V_WMMA_F32_16X16X4_F32 -------------
// Block tile 128(M) x 64(N), K panels of 32.  8 waves; each wave owns a
// 32x32 patch (2x2 accumulators).  ISA f32 fragment layouts:
//   A 16x4:  VGPR0 = K0 (lanes 0-15) / K2 (lanes 16-31); VGPR1 = K1 / K3
//   B 4x16:  same split -> one v2f (ds_load_b64) per fragment.
__global__ __launch_bounds__(256) void neumann_gemm(
    const float* __restrict__ Amat, const float* __restrict__ X,
    float* __restrict__ Y, int addI) {
  __shared__ v2f AsP[16][ASTRIDE];   // AsP[k/2][m] = {At[m][k], At[m][k+1]}
  __shared__ v2f BsP[16][BSTRIDE];   // BsP[k/2][n] = {X[k][n],  X[k+1][n]}
  const int tid  = threadIdx.x;
  const int lane = tid & 31;
  const int wave = tid >> 5;
  const int wm = (wave >> 1) * 32;
  const int wn = (wave & 1) * 32;
  const int blockM = blockIdx.y * 128;
  const int blockN = blockIdx.x * 64;
  const int lm = lane & 15;
  const int kp = (lane < 16) ? 0 : 1;   // pair offset: lanes 16-31 take K+2,K+3

  v8f acc[2][2] = {};

  for (int kk = 0; kk < MATN; kk += 32) {
    // stage At tile (128 x 32) into pair layout
    int f = tid * 4;
    #pragma unroll
    for (int p = 0; p < 4; ++p, f += 1024) {
      int m = f >> 5, k = f & 31;
      float4 v = *(const float4*)(Amat + (size_t)(blockM + m) * MATN + kk + k);
      v2f lo; lo.x = v.x; lo.y = v.y;
      v2f hi; hi.x = v.z; hi.y = v.w;
      AsP[(k >> 1) + 0][m] = lo;
      AsP[(k >> 1) + 1][m] = hi;
    }
    // stage X tile (32 x 64) into pair layout (component writes)
    f = tid * 4;
    #pragma unroll
    for (int p = 0; p < 2; ++p, f += 1024) {
      int k = f >> 6, n = f & 63;
      float4 v = *(const float4*)(X + (size_t)(kk + k) * MATN + blockN + n);
      int k2 = k >> 1, w = k & 1;
      BsP[k2][n + 0][w] = v.x;
      BsP[k2][n + 1][w] = v.y;
      BsP[k2][n + 2][w] = v.z;
      BsP[k2][n + 3][w] = v.w;
    }
    __syncthreads();

    #pragma unroll
    for (int k2 = 0; k2 < 16; k2 += 2) {    // k2 = k/2, K step 4
      v2f a[2], b[2];
      #pragma unroll
      for (int t = 0; t < 2; ++t) {
        a[t] = AsP[k2 + kp][wm + t * 16 + lm];
        b[t] = BsP[k2 + kp][wn + t * 16 + lm];
      }
      #pragma unroll
      for (int i = 0; i < 2; ++i)
        #pragma unroll
        for (int j = 0; j < 2; ++j)
          acc[i][j] = __builtin_amdgcn_wmma_f32_16x16x4_f32(
              false, a[i], false, b[j], (short)0, acc[i][j], false, false);
    }
    __syncthreads();
  }

  // epilogue: C/D layout -> lanes 0-15: M=r, lanes 16-31: M=r+8
  const int rowHalf = (lane < 16) ? 0 : 8;
  #pragma unroll
  for (int i = 0; i < 2; ++i)
    #pragma unroll
    for (int j = 0; j < 2; ++j)
      #pragma unroll
      for (int r = 0; r < 8; ++r) {
        int grow = blockM + wm + i * 16 + r + rowHalf;
        int gcol = blockN + wn + j * 16 + lm;
        float v = acc[i][j][r];
        if (addI && grow == gcol) v += 1.0f;
        Y[(size_t)grow * MATN + gcol] = v;
      }
}

// ---------------- z_b = R[idx_b, idx_b] @ sigma_b --------------------------
// One workgroup per batch.  sub (128x128) gathered from the L2-resident
// inverse in K-chunks of 32 into the pair layout; sigma chunk staged
// alongside.  Same 8-wave / 2x2-accumulator WMMA tiling (M=128,N=64,K=128).
__global__ __launch_bounds__(256) void causal_batch_kernel(
    const float* __restrict__ R, const int* __restrict__ nodeIdx,
    const float* __restrict__ sigma, float* __restrict__ z) {
  __shared__ v2f AsP[16][ASTRIDE];   // AsP[k/2][m] = {sub[m][kk+k], sub[m][kk+k+1]}
  __shared__ v2f BsP[16][BSTRIDE];   // BsP[k/2][d] = {sig[kk+k][d], sig[kk+k+1][d]}
  __shared__ int idx[NPER];

  const int b    = blockIdx.x;
  const int tid  = threadIdx.x;
  const int lane = tid & 31;
  const int wave = tid >> 5;
  const int wm = (wave >> 1) * 32;
  const int wn = (wave & 1) * 32;
  const int lm = lane & 15;
  const int kp = (lane < 16) ? 0 : 1;

  if (tid < NPER) idx[tid] = nodeIdx[b * NPER + tid];
  __syncthreads();

  const float* sigB = sigma + (size_t)b * NPER * DDIM;
  v8f acc[2][2] = {};

  for (int kk = 0; kk < NPER; kk += 32) {
    // gather sub chunk; per wave: k is constant, m runs across lanes so the
    // b32 LDS writes hit 32 distinct banks (conflict-free)
    #pragma unroll
    for (int e = 0; e < 16; ++e) {
      int flat = tid + e * 256;          // 4096 = 32(k) x 128(m)
      int m = flat & 127, k = flat >> 7;
      float val = R[(size_t)idx[m] * MATN + idx[kk + k]];
      AsP[k >> 1][m][k & 1] = val;
    }
    // stage sigma chunk (32 x 64)
    int f = tid * 4;
    #pragma unroll
    for (int p = 0; p < 2; ++p, f += 1024) {
      int k = f >> 6, n = f & 63;
      float4 v = *(const float4*)(sigB + (size_t)(kk + k) * DDIM + n);
      int k2 = k >> 1, w = k & 1;
      BsP[k2][n + 0][w] = v.x;
      BsP[k2][n + 1][w] = v.y;
      BsP[k2][n + 2][w] = v.z;
      BsP[k2][n + 3][w] = v.w;
    }
    __syncthreads();

    #pragma unroll
    for (int k2 = 0; k2 < 16; k2 += 2) {
      v2f a[2], bf[2];
      #pragma unroll
      for (int t = 0; t < 2; ++t) {
        a[t]  = AsP[k2 + kp][wm + t * 16 + lm];
        bf[t] = BsP[k2 + kp][wn + t * 16 + lm];
      }
      #pragma unroll
      for (int i = 0; i < 2; ++i)
        #pragma unroll
        for (int j = 0; j < 2; ++j)
          acc[i][j] = __builtin_amdgcn_wmma_f32_16x16x4_f32(
              false, a[i], false, bf[j], (short)0, acc[i][j], false, false);
    }
    __syncthreads();
  }

  const int rowHalf = (lane < 16) ? 0 : 8;
  #pragma unroll
  for (int i = 0; i < 2; ++i)
    #pragma unroll
    for (int j = 0; j < 2; ++j)
      #pragma unroll
      for (int r = 0; r < 8; ++r) {
        int grow = wm + i * 16 + r + rowHalf;      // 0..127
        int gcol = wn + j * 16 + lm;               // 0..63
        z[((size_t)b * NPER + grow) * DDIM + gcol] = acc[i][j][r];
      }
}

// ---------------------------------------------------------------------------
extern "C" void kernel_launch(void* const* d_in, const int* in_sizes, int n_in,
                              void* d_out, int out_size, void* d_ws, size_t ws_size,
                              hipStream_t stream) {
  const float* A      = (const float*)d_in[0];
  const float* mu     = (const float*)d_in[1];
  const float* logstd = (const float*)d_in[2];
  const float* eps    = (const float*)d_in[3];
  const int*   nodeIdx = (const int*)d_in[4];

  float* sigma = (float*)d_out;              // first output
  float* z     = sigma + SIG_ELEMS;          // second output

  float* At = (float*)d_ws;                  // 4MB
  float* Xa = At + (size_t)MATN * MATN;      // 4MB
  float* Xb = Xa + (size_t)MATN * MATN;      // 4MB

  // 1) sigma (elementwise, float4-vectorized)
  {
    int n4 = SIG_ELEMS / 4;
    sigma_kernel<<<(n4 + 255) / 256, 256, 0, stream>>>(mu, logstd, eps, sigma, n4);
  }

  // 2) At = A^T, X0 = I + A^T
  {
    dim3 grid(MATN / 32, MATN / 32), block(32, 8);
    transpose_init_kernel<<<grid, block, 0, stream>>>(A, At, Xa);
  }

  // 3) Neumann/Horner: X <- I + At*X, 16 iterations -> inv(I - A^T) to fp32 eps
  {
    dim3 grid(MATN / 64, MATN / 128);
    float* cur = Xa;
    float* nxt = Xb;
    for (int it = 0; it < 16; ++it) {
      neumann_gemm<<<grid, 256, 0, stream>>>(At, cur, nxt, 1);
      float* t = cur; cur = nxt; nxt = t;
    }
    // after an even number of swaps the result is back in Xa (== cur)
    // 4) batched gather + matmul
    causal_batch_kernel<<<NBATCH, 256, 0, stream>>>(cur, nodeIdx, sigma, z);
  }
}